// ICUSTOMIntGELU_65687229826222
// MI455X (gfx1250) — compile-verified
//
#include <hip/hip_runtime.h>
#include <cmath>

// ---------------------------------------------------------------------------
// IntGELU (piecewise-linear integer GELU), CDNA5 / gfx1250.
//
// Memory-bound elementwise op: 67.1M fp32 in + 67.1M fp32 out = ~537 MB
// -> ~23 us at 23.3 TB/s. No matrix structure (WMMA N/A); the CDNA5 feature
// that fits is the async global->LDS copy engine (ASYNCcnt): triple-buffered
// staging per wave overlaps the HBM stream with the per-element VALU work
// (IEEE divide + binary-tree bucketize + LDS coeff gather + FMA) without
// holding in-flight loads in VGPRs. Output uses non-temporal b128 stores.
// ---------------------------------------------------------------------------

#define SEGS 16
#define TPB  256
#define NPTS 10000
#define DEPTH 3   // async pipeline depth (LDS tiles in flight per wave)

typedef float __attribute__((ext_vector_type(4))) v4f;   // NT-store friendly

struct TabArgs {
    float lo[SEGS];   // segment lower bounds (exact: -5 + 0.625*k)
    float c1[SEGS];   // fitted slope per segment (float32)
    float c0[SEGS];   // fitted intercept per segment (float32)
};

// ---- CDNA5 async global->LDS copy (tracked by ASYNCcnt) --------------------
__device__ __forceinline__ void async_copy_b128(unsigned int lds_byte_addr,
                                                unsigned long long gaddr) {
    // VDST = LDS byte address VGPR, VADDR = 64-bit global address, saddr off.
    asm volatile("global_load_async_to_lds_b128 %0, %1, off"
                 :
                 : "v"(lds_byte_addr), "v"(gaddr)
                 : "memory");
}

// seg = searchsorted(lo_i[1:16], xi, side='right')  (count of t[k] <= xi),
// computed as a 4-level branchless binary-select tree over sorted pivots.
// tlo[i] holds t[i+1] = lo_i[i+1], i = 0..14.
__device__ __forceinline__ int bucketize(float xi, const float* tlo) {
    const bool b3 = (xi >= tlo[7]);                       // t[8]
    const float p2 = b3 ? tlo[11] : tlo[3];               // t[m+4], m in {0,8}
    const bool b2 = (xi >= p2);
    const float p1 = b2 ? (b3 ? tlo[13] : tlo[5])         // t[m+2], m in {0,4,8,12}
                        : (b3 ? tlo[9]  : tlo[1]);
    const bool b1 = (xi >= p1);
    const float r0 = b3 ? tlo[8]  : tlo[0];               // t[m+1], m in {0,2,..,14}
    const float r1 = b3 ? tlo[10] : tlo[2];
    const float r2 = b3 ? tlo[12] : tlo[4];
    const float r3 = b3 ? tlo[14] : tlo[6];
    const float p0 = b1 ? (b2 ? r3 : r1) : (b2 ? r2 : r0);
    const bool b0 = (xi >= p0);
    return (b3 ? 8 : 0) | (b2 ? 4 : 0) | (b1 ? 2 : 0) | (b0 ? 1 : 0);
}

__device__ __forceinline__ float eval_one(float x, float s,
                                          const float* tlo,      // regs
                                          const float2* cofs) {  // LDS table
    float xi = floorf(x / s);                 // matches jnp.floor(x / s), IEEE div
    float2 c = cofs[bucketize(xi, tlo)];      // ds_load_b64 gather
    return fmaf(c.x, xi, c.y) * 0x1p-24f;     // r / 2^24
}

__global__ __launch_bounds__(TPB)
void intgelu_async_kernel(const float* __restrict__ x,
                          const float* __restrict__ sf,
                          float* __restrict__ out,
                          unsigned int n, TabArgs tab) {
    __shared__ float  sh_tlo[SEGS];
    __shared__ float2 sh_c[SEGS];
    __shared__ float4 sh_x[DEPTH][TPB];       // triple buffer (12 KB)

    const float s   = sf[0];                  // wave-uniform scalar
    const int   tid = threadIdx.x;

    // Quantize the host-fitted table with runtime s:
    //   lo_i = floor(lo/s); int_c1 = floor(c1*s*2^24); int_c0 = floor(c0*2^24)
    if (tid < SEGS) {
        sh_tlo[tid] = floorf(tab.lo[tid] / s);
        sh_c[tid]   = make_float2(floorf(tab.c1[tid] * s * 16777216.0f),
                                  floorf(tab.c0[tid] * 16777216.0f));
    }
    __syncthreads();

    float tlo[SEGS - 1];
#pragma unroll
    for (int k = 0; k < SEGS - 1; ++k) tlo[k] = sh_tlo[k + 1];

    const unsigned int n4 = n >> 2;           // n < 2^31 (int in_sizes) -> fits u32
    const float4* __restrict__ x4   = (const float4*)x;
    v4f* __restrict__          out4 = (v4f*)out;

    const unsigned int stride = (unsigned int)gridDim.x * TPB;
    unsigned int idx = (unsigned int)blockIdx.x * TPB + (unsigned int)tid;

    unsigned int slot[DEPTH];
#pragma unroll
    for (int d = 0; d < DEPTH; ++d)
        slot[d] = (unsigned int)(size_t)&sh_x[d][tid];

    // Prologue: fill the pipeline.
#pragma unroll
    for (int d = 0; d < DEPTH; ++d)
        if (idx + (unsigned int)d * stride < n4)
            async_copy_b128(slot[d],
                (unsigned long long)(size_t)(x4 + (idx + (unsigned int)d * stride)));

    int p = 0;
    while (idx < n4) {
        // Oldest tile (idx) is complete once at most DEPTH-1 newer remain.
        if (idx + (DEPTH - 1) * stride < n4)
            asm volatile("s_wait_asynccnt 2" ::: "memory");
        else
            asm volatile("s_wait_asynccnt 0" ::: "memory");   // tail: drain

        const unsigned int s0 = slot[0], s1 = slot[1], s2 = slot[2];
        float4 v = sh_x[p][tid];              // ds_load_b128 from staged tile
        v4f y;
        y.x = eval_one(v.x, s, tlo, sh_c);
        y.y = eval_one(v.y, s, tlo, sh_c);
        y.z = eval_one(v.z, s, tlo, sh_c);
        y.w = eval_one(v.w, s, tlo, sh_c);
        __builtin_nontemporal_store(y, &out4[idx]);   // streaming NT b128 store

        // Refill buffer p (its data was just consumed; the dscnt wait that
        // produced v precedes this issue, so no LDS WAR hazard).
        const unsigned int nxt = idx + DEPTH * stride;
        if (nxt < n4) {
            const unsigned int cur = (p == 0) ? s0 : ((p == 1) ? s1 : s2);
            async_copy_b128(cur, (unsigned long long)(size_t)(x4 + nxt));
        }

        idx += stride;
        p = (p == DEPTH - 1) ? 0 : p + 1;
    }

    // Scalar tail in case n % 4 != 0 (not hit for 8*2048*4096, kept generic).
    const unsigned int base = n4 << 2;
    for (unsigned int i = base + (unsigned int)blockIdx.x * TPB + tid; i < n;
         i += stride)
        out[i] = eval_one(x[i], s, tlo, sh_c);

    // Second output: scaling_factor / 2^24
    if (blockIdx.x == 0 && tid == 0)
        out[n] = s * 0x1p-24f;
}

// ---------------------------------------------------------------------------
// Host-side replication of the reference's import-time polynomial fit:
// degree-1 least squares of exact GELU on linspace(-5, 5, 10000) per segment,
// segments = linspace(-5, 5, 17) with inclusive masks on both ends.
// Depends only on module constants -> deterministic every call.
// ---------------------------------------------------------------------------
static void fit_tab(TabArgs* t) {
    static float xs[NPTS], ys[NPTS];
    const float step = 10.0f / 9999.0f;
    for (int i = 0; i < NPTS; ++i) {
        float x = -5.0f + step * (float)i;
        if (i == NPTS - 1) x = 5.0f;
        xs[i] = x;
        double e = ::erf((double)x / 1.4142135623730951);
        ys[i] = (float)(0.5 * (double)x * (1.0 + e));
    }
    float bounds[SEGS + 1];
    for (int i = 0; i <= SEGS; ++i)
        bounds[i] = -5.0f + 0.625f * (float)i;   // exact in binary fp

    for (int sgi = 0; sgi < SEGS; ++sgi) {
        const float lo = bounds[sgi], hi = bounds[sgi + 1];
        double Sx = 0, Sy = 0, Sxx = 0, Sxy = 0;
        long   cnt = 0;
        for (int i = 0; i < NPTS; ++i) {
            if (xs[i] >= lo && xs[i] <= hi) {
                const double X = xs[i], Y = ys[i];
                Sx += X; Sy += Y; Sxx += X * X; Sxy += X * Y; ++cnt;
            }
        }
        const double det   = (double)cnt * Sxx - Sx * Sx;
        const double slope = ((double)cnt * Sxy - Sx * Sy) / det;
        const double inter = (Sy - slope * Sx) / (double)cnt;
        t->lo[sgi] = lo;
        t->c1[sgi] = (float)slope;
        t->c0[sgi] = (float)inter;
    }
}

extern "C" void kernel_launch(void* const* d_in, const int* in_sizes, int n_in,
                              void* d_out, int out_size, void* d_ws, size_t ws_size,
                              hipStream_t stream) {
    (void)n_in; (void)out_size; (void)d_ws; (void)ws_size;

    const float* x  = (const float*)d_in[0];
    const float* sf = (const float*)d_in[1];
    float*       out = (float*)d_out;
    const unsigned int n = (unsigned int)in_sizes[0];

    TabArgs tab;
    fit_tab(&tab);   // deterministic, recomputed every call (no caching)

    const unsigned int n4 = n >> 2;
    // ~32 float4 iterations per thread; 2048 blocks for the reference size.
    long long b = ((long long)n4 + (long long)TPB * 32 - 1) / ((long long)TPB * 32);
    if (b < 1) b = 1;
    if (b > 8192) b = 8192;

    intgelu_async_kernel<<<(int)b, TPB, 0, stream>>>(x, sf, out, n, tab);
}